// AttnGCN_64338610094425
// MI455X (gfx1250) — compile-verified
//
#include <hip/hip_runtime.h>
#include <math.h>

#define DNODE     66
#define NNODES    50000        // 3125 * 16 — row tiles always full
#define NEDGES    1000000
#define NEG_SLOPE 0.2f
#define EPS_BN    1e-5f

#define KPAD 68                // 66 rounded up to multiple of 4
#define NPAD 80                // 66 rounded up to multiple of 16

typedef float f32x2 __attribute__((ext_vector_type(2)));
typedef float f32x8 __attribute__((ext_vector_type(8)));

// ---------------------------------------------------------------------------
// Ordered-uint encoding of float for atomicMax-based scatter-max
// ---------------------------------------------------------------------------
__device__ __forceinline__ unsigned f2ord(float f) {
    unsigned u = __float_as_uint(f);
    return (u & 0x80000000u) ? ~u : (u | 0x80000000u);
}
__device__ __forceinline__ float ord2f(unsigned u) {
    return (u & 0x80000000u) ? __uint_as_float(u & 0x7FFFFFFFu)
                             : __uint_as_float(~u);
}
#define ORD_NEG_INF 0x007FFFFFu   // f2ord(-inf)

// ---------------------------------------------------------------------------
// WMMA fp32 GEMM: C[M,N] = A[M,66] @ B[66,N] (+bias) (optional ReLU)
//   - B staged in LDS zero-padded to 68x80 -> unconditional ds_loads
//   - M is a multiple of 16 -> no row bounds checks
//   - K hardcoded to 66, 17 fully-unrolled WMMA steps; tail masked by
//     value-selects (no exec divergence) + LDS zero padding
// One wave32 computes one 16x16 output tile via V_WMMA_F32_16X16X4_F32.
// ---------------------------------------------------------------------------
__global__ void gemm_wmma_kernel(const float* __restrict__ A,
                                 const float* __restrict__ B,
                                 const float* __restrict__ bias,
                                 float* __restrict__ C,
                                 int M, int N, int tiles_n,
                                 int relu, int has_bias) {
    __shared__ float ldsB[KPAD * NPAD];

    // Cooperative zero-padded stage of B[66 x N] into LDS [68 x 80]
    for (int i = threadIdx.x; i < KPAD * NPAD; i += 256) {
        int k = i / NPAD;
        int n = i - k * NPAD;
        ldsB[i] = (k < DNODE && n < N) ? B[k * N + n] : 0.f;
    }
    __syncthreads();

    int gw   = blockIdx.x * 8 + (threadIdx.x >> 5);
    int lane = threadIdx.x & 31;
    int tm = gw / tiles_n;
    int tn = gw - tm * tiles_n;
    if (tm * 16 >= M) return;               // wave-uniform exit

    int half = lane >> 4;                   // 0: lanes 0-15, 1: lanes 16-31
    int l16  = lane & 15;
    int row  = tm * 16 + l16;               // always < M (M multiple of 16)
    int col  = tn * 16 + l16;               // < NPAD always; < N checked at store

    // Per-lane A pointer: elements row*66 + half*2 + k0  (8-byte aligned)
    const f32x2* aptr = (const f32x2*)(A + (size_t)row * DNODE) + half;

    f32x8 acc = {0.f, 0.f, 0.f, 0.f, 0.f, 0.f, 0.f, 0.f};

#pragma unroll
    for (int k0 = 0; k0 < KPAD; k0 += 4) {
        f32x2 av = aptr[k0 >> 1];           // unconditional b64 load
        // Only the final step with half==1 (k=66,67) needs masking;
        // value select -> v_cndmask, no exec manipulation.
        bool kok = (k0 + half * 2) < DNODE;
        av.x = kok ? av.x : 0.f;
        av.y = kok ? av.y : 0.f;

        f32x2 bv;
        bv.x = ldsB[(k0 + half) * NPAD + col];
        bv.y = ldsB[(k0 + half + 2) * NPAD + col];

        acc = __builtin_amdgcn_wmma_f32_16x16x4_f32(
                  false, av, false, bv, (short)0, acc, false, false);
    }

    bool cok = col < N;
    float bc = (has_bias && cok) ? bias[col] : 0.f;
#pragma unroll
    for (int v = 0; v < 8; ++v) {
        int r = tm * 16 + half * 8 + v;     // always < M
        if (cok) {
            float val = acc[v] + bc;
            if (relu) val = fmaxf(val, 0.f);
            C[(size_t)r * N + col] = val;
        }
    }
}

// ---------------------------------------------------------------------------
// BatchNorm: per-column stats (one block per column, LDS tree reduce)
// ---------------------------------------------------------------------------
__global__ void bn_stats_kernel(const float* __restrict__ h,
                                float* __restrict__ mean,
                                float* __restrict__ var) {
    __shared__ float s1[256];
    __shared__ float s2[256];
    int d = blockIdx.x;
    int t = threadIdx.x;
    float a = 0.f, b = 0.f;
    for (int r = t; r < NNODES; r += 256) {
        float v = h[(size_t)r * DNODE + d];
        a += v; b += v * v;
    }
    s1[t] = a; s2[t] = b;
    __syncthreads();
    for (int off = 128; off; off >>= 1) {
        if (t < off) { s1[t] += s1[t + off]; s2[t] += s2[t + off]; }
        __syncthreads();
    }
    if (t == 0) {
        float m = s1[0] / (float)NNODES;
        mean[d] = m;
        var[d]  = s2[0] / (float)NNODES - m * m;   // biased variance
    }
}

__global__ void bn_apply_kernel(const float* __restrict__ h,
                                const float* __restrict__ mean,
                                const float* __restrict__ var,
                                const float* __restrict__ gamma,
                                const float* __restrict__ beta,
                                float* __restrict__ out, int n) {
    int i = blockIdx.x * blockDim.x + threadIdx.x;
    if (i >= n) return;
    int d = i % DNODE;
    out[i] = (h[i] - mean[d]) * rsqrtf(var[d] + EPS_BN) * gamma[d] + beta[d];
}

// ---------------------------------------------------------------------------
// Per-node attention scalars: a_src[n] = x[n]·att_src, a_dst[n] = x[n]·att_dst
// One wave32 per node, lanes over D, shfl_xor reduce.
// ---------------------------------------------------------------------------
__global__ void node_attn_kernel(const float* __restrict__ x,
                                 const float* __restrict__ att_s,
                                 const float* __restrict__ att_d,
                                 float* __restrict__ a_s,
                                 float* __restrict__ a_d, int N) {
    int wave = (blockIdx.x * blockDim.x + threadIdx.x) >> 5;
    int lane = threadIdx.x & 31;
    if (wave >= N) return;
    const float* xr = x + (size_t)wave * DNODE;
    float s1 = 0.f, s2 = 0.f;
    for (int d = lane; d < DNODE; d += 32) {
        float v = xr[d];
        s1 += v * att_s[d];
        s2 += v * att_d[d];
    }
    for (int off = 16; off; off >>= 1) {
        s1 += __shfl_xor(s1, off, 32);
        s2 += __shfl_xor(s2, off, 32);
    }
    if (lane == 0) { a_s[wave] = s1; a_d[wave] = s2; }
}

// ce = We · att_e  (single wave)
__global__ void dot66_kernel(const float* __restrict__ a,
                             const float* __restrict__ b,
                             float* __restrict__ out) {
    int lane = threadIdx.x;
    float s = 0.f;
    for (int d = lane; d < DNODE; d += 32) s += a[d] * b[d];
    for (int off = 16; off; off >>= 1) s += __shfl_xor(s, off, 32);
    if (lane == 0) out[0] = s;
}

// ---------------------------------------------------------------------------
// Edge kernels
// ---------------------------------------------------------------------------
__global__ void edge_alpha_kernel(const int* __restrict__ src,
                                  const int* __restrict__ dst,
                                  const float* __restrict__ ew,
                                  const float* __restrict__ a_s,
                                  const float* __restrict__ a_d,
                                  const float* __restrict__ ce,
                                  float* __restrict__ alpha,
                                  unsigned* __restrict__ mord, int E) {
    int e = blockIdx.x * blockDim.x + threadIdx.x;
    if (e >= E) return;
    int dd = dst[e];
    float al = a_s[src[e]] + a_d[dd] + ew[e] * ce[0];
    al = (al > 0.f) ? al : NEG_SLOPE * al;     // LeakyReLU
    alpha[e] = al;
    atomicMax(&mord[dd], f2ord(al));
}

__global__ void edge_exp_kernel(const int* __restrict__ dst,
                                const float* __restrict__ alpha,
                                const unsigned* __restrict__ mord,
                                float* __restrict__ exw,
                                float* __restrict__ denom, int E) {
    int e = blockIdx.x * blockDim.x + threadIdx.x;
    if (e >= E) return;
    int dd = dst[e];
    float m = ord2f(mord[dd]);
    if (!isfinite(m)) m = 0.f;                 // reference: non-finite max -> 0
    float ev = expf(alpha[e] - m);
    exw[e] = ev;
    atomicAdd(&denom[dd], ev);
}

// out[dst] += coef * x[src] ; one wave32 per edge, lanes over the 66 features.
__global__ void edge_aggregate_kernel(const int* __restrict__ src,
                                      const int* __restrict__ dst,
                                      const float* __restrict__ exw,
                                      const float* __restrict__ denom,
                                      const float* __restrict__ x,
                                      float* __restrict__ agg, int E) {
    int e    = (blockIdx.x * blockDim.x + threadIdx.x) >> 5;
    int lane = threadIdx.x & 31;
    if (e >= E) return;
    int s  = src[e];
    int dd = dst[e];
    const float* xr = x + (size_t)s * DNODE;
    float* ar       = agg + (size_t)dd * DNODE;
    __builtin_prefetch(xr + lane, 0, 3);       // global_prefetch: hide gather latency
    float coef = exw[e] / (denom[dd] + 1e-16f);
#pragma unroll
    for (int d0 = 0; d0 < DNODE; d0 += 32) {
        int d = d0 + lane;
        if (d < DNODE) atomicAdd(&ar[d], coef * xr[d]);
    }
}

__global__ void bias_relu_kernel(const float* __restrict__ agg,
                                 const float* __restrict__ b,
                                 float* __restrict__ out, int n) {
    int i = blockIdx.x * blockDim.x + threadIdx.x;
    if (i >= n) return;
    out[i] = fmaxf(agg[i] + b[i % DNODE], 0.f);
}

// ---------------------------------------------------------------------------
// Init kernels (workspace is poisoned; re-init every call)
// ---------------------------------------------------------------------------
__global__ void fill_f32_kernel(float* p, float v, int n) {
    int i = blockIdx.x * blockDim.x + threadIdx.x;
    if (i < n) p[i] = v;
}
__global__ void fill_u32_kernel(unsigned* p, unsigned v, int n) {
    int i = blockIdx.x * blockDim.x + threadIdx.x;
    if (i < n) p[i] = v;
}

// ---------------------------------------------------------------------------
// Host side
// ---------------------------------------------------------------------------
static void launch_gemm(const float* A, const float* B, const float* bias,
                        float* C, int M, int N, int relu, hipStream_t stream) {
    int tiles_m = M / 16;                      // M always a multiple of 16
    int tiles_n = (N + 15) / 16;
    int waves   = tiles_m * tiles_n;
    int blocks  = (waves + 7) / 8;             // 8 waves (256 threads) per block
    gemm_wmma_kernel<<<blocks, 256, 0, stream>>>(A, B, bias, C, M, N,
                                                 tiles_n, relu,
                                                 bias != nullptr ? 1 : 0);
}

extern "C" void kernel_launch(void* const* d_in, const int* in_sizes, int n_in,
                              void* d_out, int out_size, void* d_ws, size_t ws_size,
                              hipStream_t stream) {
    (void)in_sizes; (void)n_in; (void)out_size; (void)ws_size;

    const float* h     = (const float*)d_in[0];
    const int*   ei    = (const int*)d_in[1];
    const int*   src   = ei;
    const int*   dst   = ei + NEDGES;
    const float* ew    = (const float*)d_in[2];
    const float* gamma = (const float*)d_in[3];
    const float* beta  = (const float*)d_in[4];

    const float *W[3], *bvec[3], *attS[3], *attD[3], *We[3], *attE[3];
    for (int l = 0; l < 3; ++l) {
        int base = 5 + l * 6;
        W[l]    = (const float*)d_in[base + 0];
        bvec[l] = (const float*)d_in[base + 1];
        attS[l] = (const float*)d_in[base + 2];
        attD[l] = (const float*)d_in[base + 3];
        We[l]   = (const float*)d_in[base + 4];
        attE[l] = (const float*)d_in[base + 5];
    }
    const float *fcw[5], *fcb[5];
    for (int l = 0; l < 5; ++l) {
        fcw[l] = (const float*)d_in[23 + 2 * l];
        fcb[l] = (const float*)d_in[24 + 2 * l];
    }
    float* out = (float*)d_out;

    // ---- carve workspace (~50 MB) ----
    char* p = (char*)d_ws;
    auto carve = [&](size_t bytes) -> void* {
        void* r = (void*)p;
        p += (bytes + 255) & ~(size_t)255;
        return r;
    };
    const size_t NM = (size_t)NNODES * DNODE;
    float*    cur   = (float*)carve(NM * 4);       // current node features
    float*    xbuf  = (float*)carve(NM * 4);       // x = cur @ W
    float*    agg   = (float*)carve(NM * 4);       // scatter accumulator
    float*    a_s   = (float*)carve((size_t)NNODES * 4);
    float*    a_d   = (float*)carve((size_t)NNODES * 4);
    unsigned* mord  = (unsigned*)carve((size_t)NNODES * 4);
    float*    denom = (float*)carve((size_t)NNODES * 4);
    float*    alpha = (float*)carve((size_t)NEDGES * 4);
    float*    exw   = (float*)carve((size_t)NEDGES * 4);
    float*    meanv = (float*)carve(DNODE * 4);
    float*    varv  = (float*)carve(DNODE * 4);
    float*    ce    = (float*)carve(256);

    const int TB = 256;
    const int nmBlocks   = (int)((NM + TB - 1) / TB);
    const int nodeBlocks = (NNODES + TB - 1) / TB;
    const int edgeBlocks = (NEDGES + TB - 1) / TB;
    const int nodeWaveBlocks = (NNODES * 32 + TB - 1) / TB;           // wave-per-node
    const int edgeWaveBlocks = (int)(((size_t)NEDGES * 32 + TB - 1) / TB); // wave-per-edge

    // ---- BatchNorm ----
    bn_stats_kernel<<<DNODE, 256, 0, stream>>>(h, meanv, varv);
    bn_apply_kernel<<<nmBlocks, TB, 0, stream>>>(h, meanv, varv, gamma, beta,
                                                 cur, (int)NM);

    // ---- 3 GAT layers ----
    for (int l = 0; l < 3; ++l) {
        // x = cur @ W  (no bias inside conv)
        launch_gemm(cur, W[l], nullptr, xbuf, NNODES, DNODE, 0, stream);
        // per-node attention scalars
        node_attn_kernel<<<nodeWaveBlocks, TB, 0, stream>>>(xbuf, attS[l], attD[l],
                                                            a_s, a_d, NNODES);
        dot66_kernel<<<1, 32, 0, stream>>>(We[l], attE[l], ce);
        // init accumulators
        fill_u32_kernel<<<nodeBlocks, TB, 0, stream>>>(mord, ORD_NEG_INF, NNODES);
        fill_f32_kernel<<<nodeBlocks, TB, 0, stream>>>(denom, 0.f, NNODES);
        fill_f32_kernel<<<nmBlocks, TB, 0, stream>>>(agg, 0.f, (int)NM);
        // edge phase
        edge_alpha_kernel<<<edgeBlocks, TB, 0, stream>>>(src, dst, ew, a_s, a_d,
                                                         ce, alpha, mord, NEDGES);
        edge_exp_kernel<<<edgeBlocks, TB, 0, stream>>>(dst, alpha, mord, exw,
                                                       denom, NEDGES);
        edge_aggregate_kernel<<<edgeWaveBlocks, TB, 0, stream>>>(src, dst, exw,
                                                                 denom, xbuf,
                                                                 agg, NEDGES);
        // cur = relu(agg + b)
        bias_relu_kernel<<<nmBlocks, TB, 0, stream>>>(agg, bvec[l], cur, (int)NM);
    }

    // ---- FC stack: 4x (66->66, ReLU) then 66->22 ----
    float* a = cur;
    float* b = xbuf;
    for (int l = 0; l < 4; ++l) {
        launch_gemm(a, fcw[l], fcb[l], b, NNODES, DNODE, 1, stream);
        float* t = a; a = b; b = t;
    }
    launch_gemm(a, fcw[4], fcb[4], out, NNODES, 22, 0, stream);
}